// TurboQuantKV_8873402433779
// MI455X (gfx1250) — compile-verified
//
#include <hip/hip_runtime.h>
#include <hip/hip_bf16.h>
#include <math.h>

typedef __attribute__((ext_vector_type(2))) float v2f;
typedef __attribute__((ext_vector_type(8))) float v8f;
typedef __attribute__((ext_vector_type(4))) unsigned int u32x4;
typedef __attribute__((ext_vector_type(4))) int i32x4;
typedef __attribute__((ext_vector_type(8))) int i32x8;

#define DIMK   128      // feature dim
#define JLD    64       // JL dim
#define TM     128      // tokens per block
#define PITCH  132      // LDS row pitch (floats): 128 + 4 pad (bank-conflict free, TDM pad)
#define PPITCH 66       // LDS row pitch (floats) for 64-wide proj rows

// dq_scale = CLIP_VAL / bins / sqrt(DIM) = 3/7/sqrt(128)
#define DQ_SCALE   (3.0f / (7.0f * 11.313708498984761f))
#define SQRT_DIM   11.313708498984761f
#define CLIP_VAL   3.0f

__global__ __launch_bounds__(256)
void turboquant_kv_kernel(const float* __restrict__ kv,
                          const float* __restrict__ noise,
                          const float* __restrict__ Rm,
                          const float* __restrict__ Pm,
                          float* __restrict__ norm_out,   // [N]
                          float* __restrict__ dq_out,     // [N*64] packed int4 pairs as float
                          float* __restrict__ dqs_out,    // [1]
                          float* __restrict__ esc_out,    // [N]
                          float* __restrict__ sign_out)   // [N*8] packed sign bytes as float
{
    extern __shared__ float smem[];
    float* s_kv = smem;                 // TM x PITCH : kv tile, then kv_rot, then error
    float* s_pr = smem + TM * PITCH;    // TM x PPITCH: error_proj

    const int tid  = threadIdx.x;
    const int lane = tid & 31;
    const int wv   = tid >> 5;          // wave id 0..7
    const int m    = lane & 15;         // row (A) / col (B) within fragment
    const int kh   = lane >> 4;         // K-half selector
    const long long tok0 = (long long)blockIdx.x * TM;

    if (blockIdx.x == 0 && tid == 0) dqs_out[0] = DQ_SCALE;

    // ---------------- stage kv tile into LDS ----------------
#if __has_builtin(__builtin_amdgcn_tensor_load_to_lds)
    // Tensor Data Mover: one wave DMAs the whole 128x128 f32 tile, with
    // 4-DWORD padding inserted every 128 DWORDs -> LDS pitch of 132 floats.
    if (tid < 32) {
        const unsigned long long gaddr =
            (unsigned long long)(const void*)(kv + tok0 * DIMK);
        const unsigned lds_addr = __builtin_amdgcn_groupstaticsize(); // dyn-LDS base

        // D# group0: count=1 | lds_addr | global_addr[56:0] | type=2
        u32x4 g0;
        g0[0] = 1u;                                   // count=1, user mode, no gather
        g0[1] = lds_addr;                             // LDS byte address
        g0[2] = (unsigned)(gaddr & 0xffffffffull);    // global_addr[31:0]
        g0[3] = (unsigned)((gaddr >> 32) & 0x1ffffffull) | (2u << 30); // [56:32] | type=2

        // D# group1
        i32x8 g1;
        g1[0] = (2 << 16)      // data_size = 4 bytes
              | (1 << 20)      // pad_enable
              | (6 << 22)      // pad_interval: 128 DWORDs
              | (3 << 25);     // pad_amount:   4 DWORDs
        g1[1] = (DIMK << 16);  // tensor_dim0[15:0]=128 (bits 63:48)
        g1[2] = (TM   << 16);  // tensor_dim0[31:16]=0, tensor_dim1[15:0]=128
        g1[3] = (DIMK << 16);  // tensor_dim1[31:16]=0, tile_dim0=128
        g1[4] = TM;            // tile_dim1=128, tile_dim2=0
        g1[5] = DIMK;          // tensor_dim0_stride[31:0]=128
        g1[6] = 0;             // tensor_dim0_stride[47:32]=0, dim1_stride lo
        g1[7] = 0;             // tensor_dim1_stride hi
        i32x4 g2 = {0, 0, 0, 0};   // 2D tile: groups 2/3 unused
        i32x4 g3 = {0, 0, 0, 0};
#if defined(__clang_major__) && __clang_major__ >= 23
        i32x8 gz = {0, 0, 0, 0, 0, 0, 0, 0};
        __builtin_amdgcn_tensor_load_to_lds(g0, g1, g2, g3, gz, 0);
#else
        __builtin_amdgcn_tensor_load_to_lds(g0, g1, g2, g3, 0);
#endif
        __builtin_amdgcn_s_wait_tensorcnt(0);
    }
#else
    // Fallback: coalesced float4 staging
    {
        const float4* g = (const float4*)(kv + tok0 * DIMK);
        #pragma unroll
        for (int i = 0; i < 16; ++i) {
            int idx = tid + 256 * i;
            int row = idx >> 5;
            int c4  = idx & 31;
            float4 v = g[idx];
            float* dst = &s_kv[row * PITCH + c4 * 4];
            dst[0] = v.x; dst[1] = v.y; dst[2] = v.z; dst[3] = v.w;
        }
    }
#endif
    __syncthreads();

    // ---------------- GEMM1: kv_rot = kv @ R ----------------
    // wave wv computes output columns [16*wv, 16*wv+16) for all 8 row tiles
    v8f acc[8];
    {
        const v8f vzero = {0.f,0.f,0.f,0.f,0.f,0.f,0.f,0.f};
        #pragma unroll
        for (int rt = 0; rt < 8; ++rt) acc[rt] = vzero;

        const int ncol = wv * 16 + m;
        for (int kc = 0; kc < 32; ++kc) {
            const int k0 = kc * 4 + kh * 2;
            v2f b;
            b.x = Rm[(k0    ) * DIMK + ncol];
            b.y = Rm[(k0 + 1) * DIMK + ncol];
            // batch all 8 A-fragment loads, then issue 8 back-to-back WMMAs
            v2f a[8];
            #pragma unroll
            for (int rt = 0; rt < 8; ++rt) {
                const float* ar = &s_kv[(rt * 16 + m) * PITCH + k0];
                a[rt].x = ar[0]; a[rt].y = ar[1];
            }
            #pragma unroll
            for (int rt = 0; rt < 8; ++rt)
                acc[rt] = __builtin_amdgcn_wmma_f32_16x16x4_f32(
                    false, a[rt], false, b, (short)0, acc[rt], false, false);
        }
    }
    __syncthreads();   // all kv reads done -> safe to overwrite s_kv with kv_rot
    {
        const int col = wv * 16 + m;
        #pragma unroll
        for (int rt = 0; rt < 8; ++rt)
            #pragma unroll
            for (int r = 0; r < 8; ++r)
                s_kv[(rt * 16 + kh * 8 + r) * PITCH + col] = acc[rt][r];
    }
    __syncthreads();

    // ---------------- per-token: norm, int4 quantize, error (in place) ----------------
    if (tid < TM) {
        const long long tok = tok0 + tid;
        float* row = &s_kv[tid * PITCH];
        float ss = 0.f;
        #pragma unroll 4
        for (int j = 0; j < DIMK; ++j) { float v = row[j]; ss += v * v; }
        const float nrm = sqrtf(ss);
        norm_out[tok] = nrm;
        const float qmul = SQRT_DIM / ((nrm + 1e-8f) * CLIP_VAL);  // -> normalized
        const float dsc  = DQ_SCALE * nrm;                          // dequant scale
        #pragma unroll 4
        for (int j = 0; j < DIMK; j += 2) {
            float v0 = row[j], v1 = row[j + 1];
            float n0 = fminf(fmaxf(v0 * qmul, -1.f), 1.f);
            float n1 = fminf(fmaxf(v1 * qmul, -1.f), 1.f);
            float q0 = rintf(n0 * 7.f);                // ties-to-even == jnp.round
            float q1 = rintf(n1 * 7.f);
            int i0 = (int)q0, i1 = (int)q1;
            unsigned char pk = (unsigned char)((i0 & 15) | ((i1 & 15) << 4));
            dq_out[tok * 64 + (j >> 1)] = (float)(signed char)pk;
            row[j]     = v0 - q0 * dsc;                // error, in place
            row[j + 1] = v1 - q1 * dsc;
        }
    }
    __syncthreads();

    // ---------------- GEMM2: error_proj = error @ P ----------------
    // wave wv: fixed column tile ct = wv&3, row tiles rt = (wv>>2) + 2*i
    {
        const v8f vzero = {0.f,0.f,0.f,0.f,0.f,0.f,0.f,0.f};
        v8f acc2[4];
        #pragma unroll
        for (int i = 0; i < 4; ++i) acc2[i] = vzero;

        const int ct = wv & 3;
        const int rb = wv >> 2;
        const int ncol = ct * 16 + m;
        for (int kc = 0; kc < 32; ++kc) {
            const int k0 = kc * 4 + kh * 2;
            v2f b;
            b.x = Pm[(k0    ) * JLD + ncol];
            b.y = Pm[(k0 + 1) * JLD + ncol];
            v2f a[4];
            #pragma unroll
            for (int i = 0; i < 4; ++i) {
                const int rt = rb + 2 * i;
                const float* ar = &s_kv[(rt * 16 + m) * PITCH + k0];
                a[i].x = ar[0]; a[i].y = ar[1];
            }
            #pragma unroll
            for (int i = 0; i < 4; ++i)
                acc2[i] = __builtin_amdgcn_wmma_f32_16x16x4_f32(
                    false, a[i], false, b, (short)0, acc2[i], false, false);
        }
        const int col = ct * 16 + m;
        #pragma unroll
        for (int i = 0; i < 4; ++i) {
            const int rt = rb + 2 * i;
            #pragma unroll
            for (int r = 0; r < 8; ++r)
                s_pr[(rt * 16 + kh * 8 + r) * PPITCH + col] = acc2[i][r];
        }
    }
    __syncthreads();

    // ---------------- per-token: error_scale + sign pack ----------------
    if (tid < TM) {
        const long long tok = tok0 + tid;
        const float* prow = &s_pr[tid * PPITCH];
        float mx = 0.f;
        #pragma unroll 4
        for (int j = 0; j < JLD; ++j) mx = fmaxf(mx, fabsf(prow[j]));
        const float esc = fmaxf(mx, 1e-8f);
        esc_out[tok] = esc;
        const float rinv = 1.f / esc;
        const float* nz = noise + tok * JLD;
        #pragma unroll
        for (int b = 0; b < 8; ++b) {
            unsigned int byte = 0;
            #pragma unroll
            for (int i = 0; i < 8; ++i) {
                int j = b * 8 + i;
                float s = prow[j] * rinv - (nz[j] * 2.f - 1.f);
                if (s > 0.f) byte |= (1u << i);
            }
            sign_out[tok * 8 + b] = (float)(signed char)(unsigned char)byte;
        }
    }
}

extern "C" void kernel_launch(void* const* d_in, const int* in_sizes, int n_in,
                              void* d_out, int out_size, void* d_ws, size_t ws_size,
                              hipStream_t stream) {
    const float* kv    = (const float*)d_in[0];
    const float* noise = (const float*)d_in[1];
    const float* Rm    = (const float*)d_in[2];
    const float* Pm    = (const float*)d_in[3];

    const long long Ntok = (long long)in_sizes[0] / DIMK;   // 524288

    float* out      = (float*)d_out;
    float* norm_out = out;                       // Ntok
    float* dq_out   = norm_out + Ntok;           // Ntok*64
    float* dqs_out  = dq_out + Ntok * 64;        // 1
    float* esc_out  = dqs_out + 1;               // Ntok
    float* sign_out = esc_out + Ntok;            // Ntok*8

    const size_t shmem = (size_t)(TM * PITCH + TM * PPITCH) * sizeof(float); // ~101 KB
    (void)hipFuncSetAttribute((const void*)turboquant_kv_kernel,
                              hipFuncAttributeMaxDynamicSharedMemorySize, (int)shmem);

    dim3 grid((unsigned)(Ntok / TM));
    turboquant_kv_kernel<<<grid, 256, shmem, stream>>>(
        kv, noise, Rm, Pm, norm_out, dq_out, dqs_out, esc_out, sign_out);
}